// SideWindowFilter_72181220377220
// MI455X (gfx1250) — compile-verified
//
#include <hip/hip_runtime.h>

// Side-window filter, radius 2, 3 iterations, 24 planes of 768x768 fp32.
// Memory-bound: ~340 MB total traffic -> ~15us floor at 23.3 TB/s.
// Data movement path: async global->LDS (GLOBAL_LOAD_ASYNC_TO_LDS_B32,
// ASYNCcnt) when the builtin exists; plain loads otherwise.

#define HH 768
#define WW 768
#define NIMG 24
#define RAD 2
#define TILE 64
#define TL (TILE + 2 * RAD)   // 68
#define NTHREADS 256

#if __has_builtin(__builtin_amdgcn_global_load_async_to_lds_b32)
#define HAVE_ASYNC_LDS 1
typedef __attribute__((address_space(1))) int* swf_gptr_t;
typedef __attribute__((address_space(3))) int* swf_lptr_t;
#endif

__global__ __launch_bounds__(NTHREADS)
void swf_step_kernel(const float* __restrict__ src, float* __restrict__ dst) {
    __shared__ float tile[TL][TL];

    const int tid = threadIdx.x;
    const int x0 = blockIdx.x * TILE;
    const int y0 = blockIdx.y * TILE;
    const int img = blockIdx.z;
    const float* __restrict__ sbase = src + (size_t)img * (HH * WW);

    // ---------------- Load 68x68 tile (with halo, zero-padded) into LDS ----
    for (int idx = tid; idx < TL * TL; idx += NTHREADS) {
        const int ly = idx / TL;
        const int lx = idx - ly * TL;
        const int gy = y0 - RAD + ly;
        const int gx = x0 - RAD + lx;
        const bool inb = (gy >= 0) & (gy < HH) & (gx >= 0) & (gx < WW);
#ifdef HAVE_ASYNC_LDS
        if (inb) {
            swf_gptr_t g = (swf_gptr_t)(sbase + (size_t)gy * WW + gx);
            swf_lptr_t l = (swf_lptr_t)&tile[ly][lx];
            // async copy global -> LDS, tracked by ASYNCcnt
            __builtin_amdgcn_global_load_async_to_lds_b32(g, l, 0, 0);
        } else {
            tile[ly][lx] = 0.0f;   // zero padding (disjoint addresses from async lanes)
        }
#else
        float v = 0.0f;
        if (inb) v = sbase[(size_t)gy * WW + gx];
        tile[ly][lx] = v;
#endif
    }
#ifdef HAVE_ASYNC_LDS
#if __has_builtin(__builtin_amdgcn_s_wait_asynccnt)
    __builtin_amdgcn_s_wait_asynccnt(0);
#else
    asm volatile("s_wait_asynccnt 0" ::: "memory");
#endif
#endif
    __syncthreads();

    // ---------------- Compute: thread = one column, 16 rows ----------------
    const int cx  = tid & (TILE - 1);  // tile column 0..63 (lane-consecutive)
    const int rg  = tid >> 6;          // row group 0..3
    const int lxc = cx + RAD;          // LDS column of center
    const int base = rg * 16;          // first output row (tile coords)

    // numpy float32 rounding chain for the kernel scales
    const float s15 = 1.0f / 15.0f;
    const float s9  = 1.0f / 9.0f;
    const float s81 = s9 / 9.0f;       // NE/SE bug: (normalized 1/9) / 9

    float HL[5], HR[5], HC[5];         // sliding window of row partial sums
#pragma unroll
    for (int k = 0; k < 4; ++k) {
        const int lr = base + k;       // LDS rows base .. base+3
        float m2 = tile[lr][lxc - 2], m1 = tile[lr][lxc - 1], c0 = tile[lr][lxc];
        float p1 = tile[lr][lxc + 1], p2 = tile[lr][lxc + 2];
        HL[k] = m2 + m1 + c0;
        HR[k] = c0 + p1 + p2;
        HC[k] = c0;
    }

    float* __restrict__ dbase = dst + (size_t)img * (HH * WW);

#pragma unroll
    for (int j = 0; j < 16; ++j) {
        {   // bring in row base+j+4 (window rows ry..ry+4, center ry+2)
            const int lr = base + j + 4;
            float m2 = tile[lr][lxc - 2], m1 = tile[lr][lxc - 1], c0 = tile[lr][lxc];
            float p1 = tile[lr][lxc + 1], p2 = tile[lr][lxc + 2];
            HL[4] = m2 + m1 + c0;
            HR[4] = c0 + p1 + p2;
            HC[4] = c0;
        }

        const float sumNW = HL[0] + HL[1] + HL[2];          // top-left 3x3
        const float sumSW = HL[2] + HL[3] + HL[4];          // bottom-left 3x3
        const float sumL  = sumNW + sumSW - HL[2];          // left 5x3
        const float topR  = HR[0] + HR[1] + HR[2];
        const float botR  = HR[2] + HR[3] + HR[4];
        const float sumR  = topR + botR - HR[2];            // right 5x3
        const float topC  = HC[0] + HC[1] + HC[2];
        const float botC  = HC[2] + HC[3] + HC[4];
        const float sumU  = sumNW + topR - topC;            // top 3x5
        const float sumD  = sumSW + botR - botC;            // bottom 3x5
        const float xc    = HC[2];

        // d-channels in reference order L,R,U,D,NW,NE,SW,SE; first-min wins
        float best = sumL * s15 - xc;
        float ba   = fabsf(best);
#define SWF_CK(expr) { const float _d = (expr); const float _a = fabsf(_d); \
                       if (_a < ba) { ba = _a; best = _d; } }
        SWF_CK(sumR  * s15 - xc);   // R
        SWF_CK(sumU  * s15 - xc);   // U
        SWF_CK(sumD  * s15 - xc);   // D
        SWF_CK(sumNW * s9  - xc);   // NW
        SWF_CK(sumNW * s81 - xc);   // NE (bug: NW support, /81)
        SWF_CK(sumSW * s9  - xc);   // SW
        SWF_CK(sumSW * s81 - xc);   // SE (bug: SW support, /81)
#undef SWF_CK

        const int gy = y0 + base + j;
        dbase[(size_t)gy * WW + (x0 + cx)] = xc + best;     // coalesced per wave

        // slide window
        HL[0] = HL[1]; HL[1] = HL[2]; HL[2] = HL[3]; HL[3] = HL[4];
        HR[0] = HR[1]; HR[1] = HR[2]; HR[2] = HR[3]; HR[3] = HR[4];
        HC[0] = HC[1]; HC[1] = HC[2]; HC[2] = HC[3]; HC[3] = HC[4];
    }
}

extern "C" void kernel_launch(void* const* d_in, const int* in_sizes, int n_in,
                              void* d_out, int out_size, void* d_ws, size_t ws_size,
                              hipStream_t stream) {
    (void)in_sizes; (void)n_in; (void)out_size; (void)ws_size;
    const float* x   = (const float*)d_in[0];   // [8,3,768,768] fp32
    float*       out = (float*)d_out;           // same shape
    float*       tmp = (float*)d_ws;            // scratch plane buffer (56.6 MB)

    dim3 grid(WW / TILE, HH / TILE, NIMG);      // 12 x 12 x 24
    dim3 block(NTHREADS);

    // iteration = 3 (fixed by setup_inputs); ping-pong: in->out->ws->out
    swf_step_kernel<<<grid, block, 0, stream>>>(x,   out);
    swf_step_kernel<<<grid, block, 0, stream>>>(out, tmp);
    swf_step_kernel<<<grid, block, 0, stream>>>(tmp, out);
}